// FastFlexAddAttention_8847632630478
// MI455X (gfx1250) — compile-verified
//
#include <hip/hip_runtime.h>
#include <hip/hip_bf16.h>
#include <math.h>

// ---------------------------------------------------------------------------
// out[b] = ( sum_i w_i * x_i ) @ W_proj + b_proj        (since sum_i w_i = 1)
// Two streaming passes over x (2 * 268 MB ~ 23 us @ 23.3 TB/s) + tiny WMMA GEMM.
// One workgroup per segment (segment_ids sorted) -> no atomics, deterministic.
// ---------------------------------------------------------------------------

typedef __attribute__((ext_vector_type(2))) float v2f;
typedef __attribute__((ext_vector_type(8))) float v8f;

// Pass 1: scores + per-segment raw stats. Block = 256 threads = 8 waves.
__global__ void ffa_score_kernel(const float* __restrict__ x,
                                 const float* __restrict__ Wscore,
                                 const float* __restrict__ b_score,
                                 const int*   __restrict__ seg_ids,
                                 int N,
                                 float* __restrict__ score,
                                 float* __restrict__ sum_a,
                                 float* __restrict__ sumsq_a,
                                 float* __restrict__ smaxraw_a,
                                 int*   __restrict__ start_a,
                                 int*   __restrict__ cnt_a) {
    const int seg = blockIdx.x;
    // lower_bound(seg) and lower_bound(seg+1) on sorted segment_ids
    int lo = 0, hi = N;
    while (lo < hi) { int mid = (lo + hi) >> 1; if (seg_ids[mid] < seg) lo = mid + 1; else hi = mid; }
    const int start = lo;
    hi = N;
    while (lo < hi) { int mid = (lo + hi) >> 1; if (seg_ids[mid] < seg + 1) lo = mid + 1; else hi = mid; }
    const int end = lo;

    const int tid = threadIdx.x, wave = tid >> 5, lane = tid & 31;
    const float4 wv = ((const float4*)Wscore)[lane];   // lane owns cols 4*lane..4*lane+3
    const float  bs = b_score[0];
    const float4* __restrict__ x4 = (const float4*)x;

    float lsum = 0.f, lsq = 0.f, lmax = -INFINITY;     // valid on lane 0
    for (int r = start + wave; r < end; r += 8) {
        float4 v = x4[r * 32 + lane];                  // 512B per wave, coalesced
        float p = v.x * wv.x + v.y * wv.y + v.z * wv.z + v.w * wv.w;
        #pragma unroll
        for (int m = 16; m > 0; m >>= 1) p += __shfl_xor(p, m, 32);
        float s = p + bs;
        if (lane == 0) { score[r] = s; lsum += s; lsq += s * s; lmax = fmaxf(lmax, s); }
    }
    __shared__ float ssum[8], ssq[8], smx[8];
    if (lane == 0) { ssum[wave] = lsum; ssq[wave] = lsq; smx[wave] = lmax; }
    __syncthreads();
    if (tid == 0) {
        float S = 0.f, Q = 0.f, M = -INFINITY;
        #pragma unroll
        for (int w = 0; w < 8; ++w) { S += ssum[w]; Q += ssq[w]; M = fmaxf(M, smx[w]); }
        sum_a[seg] = S; sumsq_a[seg] = Q; smaxraw_a[seg] = M;
        start_a[seg] = start; cnt_a[seg] = end - start;
    }
}

// Pass 2: tiny finalize of segment stats.
__global__ void ffa_finalize_kernel(const float* __restrict__ sum_a,
                                    const float* __restrict__ sumsq_a,
                                    const float* __restrict__ smaxraw_a,
                                    const int*   __restrict__ cnt_a,
                                    int B,
                                    float* __restrict__ mean_a,
                                    float* __restrict__ inv_a,
                                    float* __restrict__ smaxstd_a) {
    int b = blockIdx.x * blockDim.x + threadIdx.x;
    if (b >= B) return;
    float c    = (float)cnt_a[b];
    float mean = sum_a[b] / c;
    float var  = (sumsq_a[b] - c * mean * mean) / (c - 1.0f);
    var = fmaxf(var, 0.0f);
    float inv  = 1.0f / (sqrtf(var) + 1e-7f);
    mean_a[b]      = mean;
    inv_a[b]       = inv;
    smaxstd_a[b]   = (smaxraw_a[b] - mean) * inv;  // std>0 => max(score) maps to max(s)
}

// Pass 3: softmax denom + weighted pooling of x. Block per segment, no atomics.
__global__ void ffa_pool_kernel(const float* __restrict__ x,
                                const float* __restrict__ score,
                                const float* __restrict__ mean_a,
                                const float* __restrict__ inv_a,
                                const float* __restrict__ smaxstd_a,
                                const int*   __restrict__ start_a,
                                const int*   __restrict__ cnt_a,
                                float* __restrict__ pooled) {
    const int seg   = blockIdx.x;
    const int start = start_a[seg];
    const int end   = start + cnt_a[seg];
    const float mean = mean_a[seg], inv = inv_a[seg], smax = smaxstd_a[seg];
    const int tid = threadIdx.x, wave = tid >> 5, lane = tid & 31;

    __shared__ float red[256];
    float local = 0.f;
    for (int r = start + tid; r < end; r += 256)
        local += __expf((score[r] - mean) * inv - smax);
    red[tid] = local;
    __syncthreads();
    #pragma unroll
    for (int s = 128; s > 0; s >>= 1) {          // deterministic tree reduce
        if (tid < s) red[tid] += red[tid + s];
        __syncthreads();
    }
    const float denom = red[0];
    __syncthreads();

    const float4* __restrict__ x4 = (const float4*)x;
    float4 acc = make_float4(0.f, 0.f, 0.f, 0.f);
    for (int r = start + wave; r < end; r += 8) {
        float w = __expf((score[r] - mean) * inv - smax) / denom; // uniform 4B load/row
        float4 v = x4[r * 32 + lane];
        acc.x += w * v.x; acc.y += w * v.y; acc.z += w * v.z; acc.w += w * v.w;
    }
    __shared__ float accs[8][128];
    ((float4*)accs[wave])[lane] = acc;
    __syncthreads();
    if (tid < 128) {
        float s = 0.f;
        #pragma unroll
        for (int w = 0; w < 8; ++w) s += accs[w][tid];
        pooled[seg * 128 + tid] = s;
    }
}

// Pass 4: out = pooled @ W_proj + b_proj via V_WMMA_F32_16X16X4_F32.
// One wave per 16x16 output tile; 32 WMMAs over K=128.
__global__ void ffa_gemm_kernel(const float* __restrict__ pooled,
                                const float* __restrict__ Wp,
                                const float* __restrict__ bp,
                                float* __restrict__ out) {
    const int nt   = blockIdx.x;       // N tile: 0..7
    const int mt   = blockIdx.y;       // M tile: 0..B/16-1
    const int lane = threadIdx.x;      // wave32
    const int half = lane >> 4;        // 0: K pair {0,1}; 1: K pair {2,3}
    const int idx  = lane & 15;        // A: row m; B: col n

    const float* __restrict__ arow = pooled + (mt * 16 + idx) * 128;
    const float* __restrict__ bcol = Wp + nt * 16 + idx;

    v8f c = {0.f, 0.f, 0.f, 0.f, 0.f, 0.f, 0.f, 0.f};
    #pragma unroll
    for (int k0 = 0; k0 < 128; k0 += 4) {
        const int kk = k0 + 2 * half;
        v2f a; a.x = arow[kk];            a.y = arow[kk + 1];            // A 16x4 layout
        v2f b; b.x = bcol[kk * 128];      b.y = bcol[(kk + 1) * 128];    // B 4x16 layout
        c = __builtin_amdgcn_wmma_f32_16x16x4_f32(
                false, a, false, b, (short)0, c, false, false);
    }
    const int n = nt * 16 + idx;
    const float bias = bp[n];
    #pragma unroll
    for (int r = 0; r < 8; ++r) {                 // C: VGPR r -> rows r, r+8
        const int m = mt * 16 + r + 8 * half;
        out[m * 128 + n] = c[r] + bias;
    }
}

extern "C" void kernel_launch(void* const* d_in, const int* in_sizes, int n_in,
                              void* d_out, int out_size, void* d_ws, size_t ws_size,
                              hipStream_t stream) {
    const float* x       = (const float*)d_in[0];
    const float* W_proj  = (const float*)d_in[1];
    const float* b_proj  = (const float*)d_in[2];
    const float* W_score = (const float*)d_in[3];
    const float* b_score = (const float*)d_in[4];
    const int*   seg_ids = (const int*)d_in[5];
    (void)n_in; (void)ws_size;

    const int N = in_sizes[5];          // rows
    const int B = out_size / 128;       // segments (d_out = [B,128] f32)

    // workspace layout (floats)
    float* score   = (float*)d_ws;          // N
    float* sum_a   = score + N;             // B
    float* sumsq_a = sum_a + B;             // B
    float* smaxr_a = sumsq_a + B;           // B
    float* mean_a  = smaxr_a + B;           // B
    float* inv_a   = mean_a + B;            // B
    float* smaxs_a = inv_a + B;             // B
    int*   start_a = (int*)(smaxs_a + B);   // B
    int*   cnt_a   = start_a + B;           // B
    float* pooled  = (float*)(cnt_a + B);   // B*128

    ffa_score_kernel<<<B, 256, 0, stream>>>(x, W_score, b_score, seg_ids, N,
                                            score, sum_a, sumsq_a, smaxr_a,
                                            start_a, cnt_a);
    ffa_finalize_kernel<<<(B + 255) / 256, 256, 0, stream>>>(
        sum_a, sumsq_a, smaxr_a, cnt_a, B, mean_a, inv_a, smaxs_a);
    ffa_pool_kernel<<<B, 256, 0, stream>>>(x, score, mean_a, inv_a, smaxs_a,
                                           start_a, cnt_a, pooled);
    ffa_gemm_kernel<<<dim3(8, B / 16), 32, 0, stream>>>(pooled, W_proj, b_proj,
                                                        (float*)d_out);
}